// GCNNet_12137577579001
// MI455X (gfx1250) — compile-verified
//
#include <hip/hip_runtime.h>
#include <math.h>

#define NN 50000
#define NE 500000
#define NG 256
#define FX 78

typedef float v2f __attribute__((ext_vector_type(2)));
typedef float v8f __attribute__((ext_vector_type(8)));

// ---------------- CSR construction ----------------

__global__ void k_zero_u32(unsigned* p, int n) {
  int i = blockIdx.x * 256 + threadIdx.x;
  if (i < n) p[i] = 0u;
}

__global__ void k_count(const int* __restrict__ dst, unsigned* __restrict__ counts, int E) {
  int e = blockIdx.x * 256 + threadIdx.x;
  if (e < E) atomicAdd(&counts[dst[e]], 1u);
}

__global__ void k_scan_block(const unsigned* __restrict__ in, unsigned* __restrict__ excl,
                             unsigned* __restrict__ bsums, int n) {
  __shared__ unsigned sh[256];
  int i = blockIdx.x * 256 + threadIdx.x;
  unsigned v = (i < n) ? in[i] : 0u;
  sh[threadIdx.x] = v;
  __syncthreads();
  for (int off = 1; off < 256; off <<= 1) {
    unsigned t = (threadIdx.x >= (unsigned)off) ? sh[threadIdx.x - off] : 0u;
    __syncthreads();
    sh[threadIdx.x] += t;
    __syncthreads();
  }
  if (i < n) excl[i] = sh[threadIdx.x] - v;
  if (threadIdx.x == 255) bsums[blockIdx.x] = sh[255];
}

__global__ void k_scan_sums(unsigned* bsums, int nb) {
  __shared__ unsigned sh[256];
  unsigned v = (threadIdx.x < (unsigned)nb) ? bsums[threadIdx.x] : 0u;
  sh[threadIdx.x] = v;
  __syncthreads();
  for (int off = 1; off < 256; off <<= 1) {
    unsigned t = (threadIdx.x >= (unsigned)off) ? sh[threadIdx.x - off] : 0u;
    __syncthreads();
    sh[threadIdx.x] += t;
    __syncthreads();
  }
  if (threadIdx.x < (unsigned)nb) bsums[threadIdx.x] = sh[threadIdx.x] - v;
}

__global__ void k_scan_add(unsigned* __restrict__ offs, const unsigned* __restrict__ bscan,
                           unsigned* __restrict__ pos, int n, unsigned total) {
  int i = blockIdx.x * 256 + threadIdx.x;
  if (i < n) {
    unsigned o = offs[i] + bscan[blockIdx.x];
    offs[i] = o;
    pos[i] = o;
  }
  if (i == 0) offs[n] = total;
}

__global__ void k_dinv(const unsigned* __restrict__ counts, float* __restrict__ dinv, int n) {
  int i = blockIdx.x * 256 + threadIdx.x;
  if (i < n) dinv[i] = rsqrtf((float)counts[i] + 1.0f);
}

__global__ void k_fill(const int* __restrict__ src, const int* __restrict__ dst,
                       const float* __restrict__ dinv, unsigned* __restrict__ pos,
                       int* __restrict__ csr_src, float* __restrict__ csr_w, int E) {
  int e = blockIdx.x * 256 + threadIdx.x;
  if (e >= E) return;
  int s = src[e], d = dst[e];
  unsigned slot = atomicAdd(&pos[d], 1u);
  csr_src[slot] = s;
  csr_w[slot] = dinv[s] * dinv[d];
}

// ---------------- padding ----------------

// plain zero-pad (row-major), used for the A operand (X)
__global__ void k_pad_mat(const float* __restrict__ S, float* __restrict__ D,
                          int K, int N, int Kp, int Np) {
  int i = blockIdx.x * 256 + threadIdx.x;
  long tot = (long)Kp * Np;
  if (i >= tot) return;
  int r = i / Np, c = i % Np;
  D[i] = (r < K && c < N) ? S[(long)r * N + c] : 0.0f;
}

// zero-pad + K-pair swizzle for B operands:  D[((r>>1)*Np + c)*2 + (r&1)] = S[r][c]
// -> a lane's B fragment {W[kr][c], W[kr+1][c]} is one aligned 8-byte load.
__global__ void k_pad_swz(const float* __restrict__ S, float* __restrict__ D,
                          int K, int N, int Kp, int Np) {
  int i = blockIdx.x * 256 + threadIdx.x;
  long tot = (long)Kp * Np;
  if (i >= tot) return;
  int r = i / Np, c = i % Np;
  float v = (r < K && c < N) ? S[(long)r * N + c] : 0.0f;
  D[((size_t)(r >> 1) * Np + c) * 2 + (r & 1)] = v;
}

__global__ void k_pad_vec(const float* __restrict__ S, float* __restrict__ D, int N, int Np) {
  int i = blockIdx.x * 256 + threadIdx.x;
  if (i < Np) D[i] = (i < N) ? S[i] : 0.0f;
}

// ---------------- WMMA GEMM:  out[M,Np] = X[M,Kp] @ W[Kp,Np] (+bias)(+relu) ----------------
// W in K-pair swizzled layout (k_pad_swz). 4 waves/block, each wave one 16x(16*NTA) tile.
// Kp % 4 == 0, Np % 16 == 0, M % 16 == 0.
// NTA is a compile-time constant so all accumulator indices are constant (no v_movrel).
template <int NTA>
__device__ __forceinline__ void gemm_body(const float* __restrict__ xr,
                                          const float* __restrict__ wb, size_t wstep2,
                                          const float* __restrict__ bias,
                                          float* __restrict__ out,
                                          int Kp, int Np, int relu,
                                          int mBase, int nBase0, int half, int l16) {
  v8f acc[NTA] = {};
#pragma unroll 2
  for (int k = 0; k < Kp; k += 4) {
    v2f a = *(const v2f*)(xr + k);
    const float* wk = wb + (size_t)(k >> 1) * wstep2;
#pragma unroll
    for (int t = 0; t < NTA; ++t) {
      v2f b = *(const v2f*)(wk + 32 * t);
      acc[t] = __builtin_amdgcn_wmma_f32_16x16x4_f32(false, a, false, b, (short)0,
                                                     acc[t], false, false);
    }
  }
#pragma unroll
  for (int t = 0; t < NTA; ++t) {
    int col = nBase0 + 16 * t + l16;
    float bv = bias ? bias[col] : 0.0f;
#pragma unroll
    for (int i = 0; i < 8; ++i) {
      int r = mBase + i + 8 * half;              // C/D layout: vgpr i -> M=i (+8 for hi half)
      float v = acc[t][i] + bv;
      if (relu) v = fmaxf(v, 0.0f);
      out[(size_t)r * Np + col] = v;
    }
  }
}

__global__ void __launch_bounds__(128)
k_gemm(const float* __restrict__ X, const float* __restrict__ W,
       const float* __restrict__ bias, float* __restrict__ out,
       int Mtiles, int Kp, int Np, int relu) {
  const int NT = 4;
  int lane = threadIdx.x & 31;
  int wave = threadIdx.x >> 5;
  int mTile = blockIdx.x * 4 + wave;
  if (mTile >= Mtiles) return;                   // wave-uniform
  int half = lane >> 4;                          // 0: K {k,k+1}/rows M0-7 ; 1: K {k+2,k+3}/rows M8-15
  int l16  = lane & 15;
  int mBase = mTile * 16;
  int nBase0 = blockIdx.y * (16 * NT);
  int ntAct = (Np - nBase0) >> 4;
  if (ntAct > NT) ntAct = NT;

  const float* __restrict__ xr = X + (size_t)(mBase + l16) * Kp + 2 * half;
  // swizzled B base for this lane: element ((k/2 + half)*Np + nBase0 + l16)*2
  const float* __restrict__ wb = W + ((size_t)half * Np + nBase0 + l16) * 2;
  const size_t wstep2 = (size_t)2 * Np;          // advance per (k>>1) increment

  switch (ntAct) {                               // wave-uniform dispatch
    case 4: gemm_body<4>(xr, wb, wstep2, bias, out, Kp, Np, relu, mBase, nBase0, half, l16); break;
    case 3: gemm_body<3>(xr, wb, wstep2, bias, out, Kp, Np, relu, mBase, nBase0, half, l16); break;
    case 2: gemm_body<2>(xr, wb, wstep2, bias, out, Kp, Np, relu, mBase, nBase0, half, l16); break;
    default: gemm_body<1>(xr, wb, wstep2, bias, out, Kp, Np, relu, mBase, nBase0, half, l16); break;
  }
}

// ---------------- GCN aggregation: out = relu( dinv^2*H[node] + sum_e w_e*H[src_e] + b ) ----
__global__ void k_agg(const float* __restrict__ H, const float* __restrict__ dinv,
                      const unsigned* __restrict__ offs, const int* __restrict__ csr_src,
                      const float* __restrict__ csr_w, const float* __restrict__ bias,
                      float* __restrict__ out, int fp) {
  __shared__ int   ssrc[128];
  __shared__ float sw[128];
  int node = blockIdx.x;
  float dn = dinv[node];
  float selfw = dn * dn;
  unsigned e0 = offs[node], e1 = offs[node + 1];
  int f0 = threadIdx.x, f1 = f0 + 128, f2 = f0 + 256;
  size_t nb = (size_t)node * fp;
  float a0 = (f0 < fp) ? selfw * H[nb + f0] : 0.0f;
  float a1 = (f1 < fp) ? selfw * H[nb + f1] : 0.0f;
  float a2 = (f2 < fp) ? selfw * H[nb + f2] : 0.0f;
  for (unsigned base = e0; base < e1; base += 128u) {
    unsigned cnt = e1 - base;
    if (cnt > 128u) cnt = 128u;
    __syncthreads();
    if (threadIdx.x < cnt) {
      ssrc[threadIdx.x] = csr_src[base + threadIdx.x];
      sw[threadIdx.x]   = csr_w[base + threadIdx.x];
    }
    __syncthreads();
    for (unsigned c = 0; c < cnt; ++c) {
      size_t sb = (size_t)ssrc[c] * fp;
      float wv = sw[c];
      if (f0 < fp) a0 += wv * H[sb + f0];
      if (f1 < fp) a1 += wv * H[sb + f1];
      if (f2 < fp) a2 += wv * H[sb + f2];
    }
  }
  if (f0 < fp) out[nb + f0] = fmaxf(a0 + bias[f0], 0.0f);
  if (f1 < fp) out[nb + f1] = fmaxf(a1 + bias[f1], 0.0f);
  if (f2 < fp) out[nb + f2] = fmaxf(a2 + bias[f2], 0.0f);
}

// ---------------- per-graph max pool (batch = (i*NG)//NN, sorted) ----------------
__global__ void k_pool(const float* __restrict__ A, float* __restrict__ G, int fp) {
  int g = blockIdx.x;
  int start = (g * NN + NG - 1) / NG;
  int end   = ((g + 1) * NN + NG - 1) / NG;
  for (int f = threadIdx.x; f < fp; f += blockDim.x) {
    float m = -3.4e38f;
    for (int i = start; i < end; ++i) m = fmaxf(m, A[(size_t)i * fp + f]);
    G[(size_t)g * fp + f] = m;
  }
}

__global__ void k_concat(const float* __restrict__ G2, const float* __restrict__ T,
                         const float* __restrict__ P, float* __restrict__ XC) {
  int i = blockIdx.x * 256 + threadIdx.x;
  if (i >= NG * 132) return;
  int g = i / 132, j = i % 132;
  float v;
  if (j < 128)      v = G2[g * 128 + j];
  else if (j == 128) v = T[g];
  else if (j == 129) v = P[g];
  else               v = 0.0f;
  XC[i] = v;
}

__global__ void k_final(const float* __restrict__ F2, const float* __restrict__ Wo,
                        const float* __restrict__ bo, float* __restrict__ out) {
  __shared__ float sh[256];
  int g = blockIdx.x;
  float s = 0.0f;
  for (int j = threadIdx.x; j < 512; j += 256) s += F2[(size_t)g * 512 + j] * Wo[j];
  sh[threadIdx.x] = s;
  __syncthreads();
  for (int off = 128; off > 0; off >>= 1) {
    if (threadIdx.x < (unsigned)off) sh[threadIdx.x] += sh[threadIdx.x + off];
    __syncthreads();
  }
  if (threadIdx.x == 0) out[g] = sh[0] + bo[0];
}

// ---------------- driver ----------------

extern "C" void kernel_launch(void* const* d_in, const int* in_sizes, int n_in,
                              void* d_out, int out_size, void* d_ws, size_t ws_size,
                              hipStream_t stream) {
  const float* x   = (const float*)d_in[0];
  const int*   ei  = (const int*)d_in[1];
  const int*   src = ei;
  const int*   dst = ei + NE;
  const float* T   = (const float*)d_in[3];
  const float* P   = (const float*)d_in[4];
  const float* W1  = (const float*)d_in[5];  const float* b1  = (const float*)d_in[6];
  const float* W2  = (const float*)d_in[7];  const float* b2  = (const float*)d_in[8];
  const float* W3  = (const float*)d_in[9];  const float* b3  = (const float*)d_in[10];
  const float* Wg1 = (const float*)d_in[11]; const float* bg1 = (const float*)d_in[12];
  const float* Wg2 = (const float*)d_in[13]; const float* bg2 = (const float*)d_in[14];
  const float* Wf1 = (const float*)d_in[15]; const float* bf1 = (const float*)d_in[16];
  const float* Wf2 = (const float*)d_in[17]; const float* bf2 = (const float*)d_in[18];
  const float* Wo  = (const float*)d_in[19]; const float* bo  = (const float*)d_in[20];
  float* out = (float*)d_out;

  char* wp = (char*)d_ws;
  auto alloc = [&](size_t bytes) -> void* {
    void* p = (void*)wp;
    wp += (bytes + 255) & ~(size_t)255;
    return p;
  };
  float* BIG0 = (float*)alloc((size_t)NN * 320 * 4);  // GEMM outputs H
  float* BIG1 = (float*)alloc((size_t)NN * 320 * 4);  // agg outputs A (Xp aliases front)
  float* Xp   = BIG1;                                 // [NN,80]; dead once H1 is computed
  float* W1p  = (float*)alloc(80 * 80 * 4);
  float* W2p  = (float*)alloc(80 * 160 * 4);
  float* W3p  = (float*)alloc(160 * 320 * 4);
  float* Wg1p = (float*)alloc(320 * 1024 * 4);
  float* Wg2s = (float*)alloc(1024 * 128 * 4);
  float* Wf1p = (float*)alloc(132 * 1024 * 4);
  float* Wf2s = (float*)alloc(1024 * 512 * 4);
  float* b1p  = (float*)alloc(80 * 4);
  float* b2p  = (float*)alloc(160 * 4);
  float* b3p  = (float*)alloc(320 * 4);
  unsigned* counts = (unsigned*)alloc((size_t)NN * 4);
  unsigned* offs   = (unsigned*)alloc((size_t)(NN + 1) * 4);
  unsigned* bsums  = (unsigned*)alloc(256 * 4);
  unsigned* pos    = (unsigned*)alloc((size_t)NN * 4);
  int*   csr_src   = (int*)alloc((size_t)NE * 4);
  float* csr_w     = (float*)alloc((size_t)NE * 4);
  float* dinv      = (float*)alloc((size_t)NN * 4);
  float* Gp  = (float*)alloc((size_t)NG * 320 * 4);
  float* G1  = (float*)alloc((size_t)NG * 1024 * 4);
  float* G2  = (float*)alloc((size_t)NG * 128 * 4);
  float* XC  = (float*)alloc((size_t)NG * 132 * 4);
  float* F1  = (float*)alloc((size_t)NG * 1024 * 4);
  float* F2  = (float*)alloc((size_t)NG * 512 * 4);

  auto cdiv = [](long a, long b) -> int { return (int)((a + b - 1) / b); };
  const int nbN = cdiv(NN, 256);  // 196
  const int MtN = NN / 16;        // 3125 node M-tiles
  const int MtG = NG / 16;        // 16 graph M-tiles

  // ---- CSR (dst-sorted adjacency) + symmetric norm weights ----
  k_zero_u32<<<nbN, 256, 0, stream>>>(counts, NN);
  k_count<<<cdiv(NE, 256), 256, 0, stream>>>(dst, counts, NE);
  k_scan_block<<<nbN, 256, 0, stream>>>(counts, offs, bsums, NN);
  k_scan_sums<<<1, 256, 0, stream>>>(bsums, nbN);
  k_scan_add<<<nbN, 256, 0, stream>>>(offs, bsums, pos, NN, (unsigned)NE);
  k_dinv<<<nbN, 256, 0, stream>>>(counts, dinv, NN);
  k_fill<<<cdiv(NE, 256), 256, 0, stream>>>(src, dst, dinv, pos, csr_src, csr_w, NE);

  // ---- zero-pad / swizzle operands for WMMA tiling ----
  k_pad_mat<<<cdiv((long)NN * 80, 256), 256, 0, stream>>>(x,   Xp,   NN, FX, NN, 80);
  k_pad_swz<<<cdiv(80L * 80, 256),    256, 0, stream>>>(W1,  W1p,  78, 78, 80, 80);
  k_pad_swz<<<cdiv(80L * 160, 256),   256, 0, stream>>>(W2,  W2p,  78, 156, 80, 160);
  k_pad_swz<<<cdiv(160L * 320, 256),  256, 0, stream>>>(W3,  W3p,  156, 312, 160, 320);
  k_pad_swz<<<cdiv(320L * 1024, 256), 256, 0, stream>>>(Wg1, Wg1p, 312, 1024, 320, 1024);
  k_pad_swz<<<cdiv(1024L * 128, 256), 256, 0, stream>>>(Wg2, Wg2s, 1024, 128, 1024, 128);
  k_pad_swz<<<cdiv(132L * 1024, 256), 256, 0, stream>>>(Wf1, Wf1p, 130, 1024, 132, 1024);
  k_pad_swz<<<cdiv(1024L * 512, 256), 256, 0, stream>>>(Wf2, Wf2s, 1024, 512, 1024, 512);
  k_pad_vec<<<1, 256, 0, stream>>>(b1, b1p, 78, 80);
  k_pad_vec<<<1, 256, 0, stream>>>(b2, b2p, 156, 160);
  k_pad_vec<<<2, 256, 0, stream>>>(b3, b3p, 312, 320);

  // ---- GCN layer 1: H1 = Xp @ W1p ; A1 = relu(agg(H1) + b1) ----
  k_gemm<<<dim3(cdiv(MtN, 4), cdiv(80 / 16, 4)), 128, 0, stream>>>(Xp, W1p, nullptr, BIG0, MtN, 80, 80, 0);
  k_agg<<<NN, 128, 0, stream>>>(BIG0, dinv, offs, csr_src, csr_w, b1p, BIG1, 80);
  // ---- GCN layer 2 ----
  k_gemm<<<dim3(cdiv(MtN, 4), cdiv(160 / 16, 4)), 128, 0, stream>>>(BIG1, W2p, nullptr, BIG0, MtN, 80, 160, 0);
  k_agg<<<NN, 128, 0, stream>>>(BIG0, dinv, offs, csr_src, csr_w, b2p, BIG1, 160);
  // ---- GCN layer 3 ----
  k_gemm<<<dim3(cdiv(MtN, 4), cdiv(320 / 16, 4)), 128, 0, stream>>>(BIG1, W3p, nullptr, BIG0, MtN, 160, 320, 0);
  k_agg<<<NN, 128, 0, stream>>>(BIG0, dinv, offs, csr_src, csr_w, b3p, BIG1, 320);

  // ---- max pool + MLP head ----
  k_pool<<<NG, 320, 0, stream>>>(BIG1, Gp, 320);
  k_gemm<<<dim3(cdiv(MtG, 4), cdiv(1024 / 16, 4)), 128, 0, stream>>>(Gp, Wg1p, bg1, G1, MtG, 320, 1024, 1);
  k_gemm<<<dim3(cdiv(MtG, 4), cdiv(128 / 16, 4)),  128, 0, stream>>>(G1, Wg2s, bg2, G2, MtG, 1024, 128, 0);
  k_concat<<<cdiv((long)NG * 132, 256), 256, 0, stream>>>(G2, T, P, XC);
  k_gemm<<<dim3(cdiv(MtG, 4), cdiv(1024 / 16, 4)), 128, 0, stream>>>(XC, Wf1p, bf1, F1, MtG, 132, 1024, 1);
  k_gemm<<<dim3(cdiv(MtG, 4), cdiv(512 / 16, 4)),  128, 0, stream>>>(F1, Wf2s, bf2, F2, MtG, 1024, 512, 1);
  k_final<<<NG, 256, 0, stream>>>(F2, Wo, bo, out);
}